// KGAT_21577915695266
// MI455X (gfx1250) — compile-verified
//
#include <hip/hip_runtime.h>
#include <hip/hip_bf16.h>
#include <math.h>

// Problem constants (match reference)
#define V_ENT 200000
#define R_REL 100
#define D_DIM 128
#define K_NBR 20
#define B_DOC 128
#define N_ENT 50
#define NK    (N_ENT * K_NBR)     // 1000
#define BN    (B_DOC * N_ENT)     // 6400
#define KD2   (2 * D_DIM)         // 256

typedef __attribute__((ext_vector_type(2))) float v2f;
typedef __attribute__((ext_vector_type(8))) float v8f;

// wave32 all-lane sum reduction
__device__ __forceinline__ float wave_sum(float v) {
    #pragma unroll
    for (int off = 16; off > 0; off >>= 1) v += __shfl_xor(v, off, 32);
    return v;
}

// ---------------------------------------------------------------------------
// Kernel 1: attention logits = relu(ee.w1 + ne.w2 + nr.w3 + b_att)
// One block per (b,n); 8 waves; each wave handles k = waveId, waveId+8, ...
// Each lane holds 4 contiguous floats of a D=128 row (32 lanes * 4 = 128).
// ---------------------------------------------------------------------------
__global__ void kgat_logits(const int* __restrict__ entity_ids,
                            const int* __restrict__ adj_entity,
                            const int* __restrict__ adj_relation,
                            const float* __restrict__ ent_emb,
                            const float* __restrict__ rel_emb,
                            const float* __restrict__ W_att,
                            const float* __restrict__ b_att,
                            float* __restrict__ logits) {
    const int bn   = blockIdx.x;            // 0..BN-1
    const int wave = threadIdx.x >> 5;      // 0..7
    const int lane = threadIdx.x & 31;

    const int eid = entity_ids[bn];

    // ee . w1  (redundant per wave; tiny, L1-resident)
    const float4 e4 = *(const float4*)(ent_emb + (size_t)eid * D_DIM + lane * 4);
    const float4 w1 = *(const float4*)(W_att + lane * 4);
    float s_e = wave_sum(e4.x * w1.x + e4.y * w1.y + e4.z * w1.z + e4.w * w1.w);

    const float4 w2 = *(const float4*)(W_att + D_DIM + lane * 4);
    const float4 w3 = *(const float4*)(W_att + 2 * D_DIM + lane * 4);
    const float bias = b_att[0];

    for (int k = wave; k < K_NBR; k += 8) {
        const int nid = adj_entity  [(size_t)eid * K_NBR + k];
        const int rid = adj_relation[(size_t)eid * K_NBR + k];
        const float4 n4 = *(const float4*)(ent_emb + (size_t)nid * D_DIM + lane * 4);
        const float4 r4 = *(const float4*)(rel_emb + (size_t)rid * D_DIM + lane * 4);
        float p = n4.x * w2.x + n4.y * w2.y + n4.z * w2.z + n4.w * w2.w
                + r4.x * w3.x + r4.y * w3.y + r4.z * w3.z + r4.w * w3.w;
        float tot = wave_sum(p) + s_e + bias;
        if (lane == 0) logits[(size_t)bn * K_NBR + k] = fmaxf(tot, 0.0f);  // relu
    }
}

// ---------------------------------------------------------------------------
// Kernel 2: softmax over the BATCH axis (faithful to reference softmax(axis=0)).
// One 128-thread block per (n,k) column; element b lives at logits[b*NK + col].
// In-place: all reads happen before writes (barrier-separated).
// ---------------------------------------------------------------------------
__global__ void kgat_softmax_b(float* __restrict__ att) {
    __shared__ float red[B_DOC];
    const int col = blockIdx.x;             // 0..NK-1
    const int t   = threadIdx.x;            // = b
    const float x = att[(size_t)t * NK + col];

    red[t] = x;
    __syncthreads();
    #pragma unroll
    for (int s = B_DOC / 2; s > 0; s >>= 1) {
        if (t < s) red[t] = fmaxf(red[t], red[t + s]);
        __syncthreads();
    }
    const float m = red[0];
    __syncthreads();

    const float e = __expf(x - m);
    red[t] = e;
    __syncthreads();
    #pragma unroll
    for (int s = B_DOC / 2; s > 0; s >>= 1) {
        if (t < s) red[t] += red[t + s];
        __syncthreads();
    }
    att[(size_t)t * NK + col] = e / red[0];
}

// ---------------------------------------------------------------------------
// Kernel 3 (fused aggregate + conv GEMM):
//   Phase 1: for 16 consecutive (b,n) rows, build agg = [ee | sum_k att*ne]
//            directly in LDS (16 x 256 f32 = 16 KB). 8 waves x 2 rows each;
//            lanes cover D=128 via float4 -> fully coalesced row gathers.
//   Phase 2: out_tile = relu(agg @ W_conv + b_conv) with V_WMMA_F32_16X16X4_F32.
//            One 16x16 output tile per wave (8 waves = all 8 N-tiles);
//            A fragments come from LDS (ds_load_b64), B from global (L1-hot).
//
// ISA fragment layouts (wave32):
//   A 16x4 : lanes 0-15 -> {K0,K1} in v0,v1 ; lanes 16-31 -> {K2,K3}
//   B 4x16 : lanes 0-15 -> rows K0,K1       ; lanes 16-31 -> rows K2,K3
//   C/D    : vgpr r, lanes 0-15 -> (M=r, N=lane); lanes 16-31 -> (M=r+8)
// EXEC is all-1s throughout phase 2 (no divergence), as WMMA requires.
// ---------------------------------------------------------------------------
__global__ void kgat_agg_conv_fused(const int* __restrict__ entity_ids,
                                    const int* __restrict__ adj_entity,
                                    const float* __restrict__ ent_emb,
                                    const float* __restrict__ att,
                                    const float* __restrict__ Wc,
                                    const float* __restrict__ bconv,
                                    float* __restrict__ out) {
    __shared__ float smem[16 * KD2];        // 16 agg rows x 256 = 16 KB
    const int m0   = blockIdx.x * 16;       // first (b,n) row of this tile
    const int wave = threadIdx.x >> 5;      // 0..7
    const int lane = threadIdx.x & 31;

    // ---- Phase 1: gather + attention-weighted sum into LDS ----
    #pragma unroll
    for (int rr = 0; rr < 2; ++rr) {
        const int row = wave * 2 + rr;      // 0..15
        const int bn  = m0 + row;
        const int eid = entity_ids[bn];

        float4 acc = make_float4(0.f, 0.f, 0.f, 0.f);
        #pragma unroll 4
        for (int k = 0; k < K_NBR; ++k) {
            const float a   = att[(size_t)bn * K_NBR + k];
            const int   nid = adj_entity[(size_t)eid * K_NBR + k];
            const float4 nv = *(const float4*)(ent_emb + (size_t)nid * D_DIM + lane * 4);
            acc.x = fmaf(a, nv.x, acc.x);
            acc.y = fmaf(a, nv.y, acc.y);
            acc.z = fmaf(a, nv.z, acc.z);
            acc.w = fmaf(a, nv.w, acc.w);
        }
        const float4 ev = *(const float4*)(ent_emb + (size_t)eid * D_DIM + lane * 4);
        *(float4*)(&smem[row * KD2 + lane * 4])         = ev;   // agg[:,0:D]   = ee
        *(float4*)(&smem[row * KD2 + D_DIM + lane * 4]) = acc;  // agg[:,D:2D]  = neighbor_att
    }
    __syncthreads();

    // ---- Phase 2: 16x16 WMMA tile per wave, K-loop over 2D=256 ----
    const int half = lane >> 4;             // 0 | 1
    const int lrow = lane & 15;
    const int n0   = wave * 16;

    v8f c = {};
    #pragma unroll 4
    for (int kb = 0; kb < KD2 / 4; ++kb) {
        const int k = kb * 4 + 2 * half;    // this half-wave's K pair
        v2f a, b;
        const float2 ap = *(const float2*)(&smem[lrow * KD2 + k]);  // ds_load_b64
        a.x = ap.x;  a.y = ap.y;
        b.x = Wc[(size_t)(k + 0) * D_DIM + n0 + lrow];
        b.y = Wc[(size_t)(k + 1) * D_DIM + n0 + lrow];
        c = __builtin_amdgcn_wmma_f32_16x16x4_f32(
                /*neg_a=*/false, a, /*neg_b=*/false, b,
                /*c_mod=*/(short)0, c, /*reuse_a=*/false, /*reuse_b=*/false);
    }

    const float bias = bconv[n0 + lrow];
    #pragma unroll
    for (int r = 0; r < 8; ++r) {
        const int m = m0 + r + 8 * half;
        out[(size_t)m * D_DIM + n0 + lrow] = fmaxf(c[r] + bias, 0.0f); // relu
    }
}

// ---------------------------------------------------------------------------
// Launch
// ---------------------------------------------------------------------------
extern "C" void kernel_launch(void* const* d_in, const int* in_sizes, int n_in,
                              void* d_out, int out_size, void* d_ws, size_t ws_size,
                              hipStream_t stream) {
    const int*   entity_ids = (const int*)  d_in[0];   // [B,N]
    const int*   adj_entity = (const int*)  d_in[1];   // [V,K]
    const int*   adj_rel    = (const int*)  d_in[2];   // [V,K]
    const float* ent_emb    = (const float*)d_in[3];   // [V,D]
    const float* rel_emb    = (const float*)d_in[4];   // [R,D]
    const float* W_att      = (const float*)d_in[5];   // [3D,1]
    const float* b_att      = (const float*)d_in[6];   // [1]
    const float* W_conv     = (const float*)d_in[7];   // [2D,D]
    const float* b_conv     = (const float*)d_in[8];   // [D]
    float* out = (float*)d_out;                        // [B,N,D]

    // Workspace: only the logits/att buffer [B*N*K] f32 (512 KB).
    float* ws_att = (float*)d_ws;

    kgat_logits        <<<BN,      256,   0, stream>>>(entity_ids, adj_entity, adj_rel,
                                                       ent_emb, rel_emb, W_att, b_att,
                                                       ws_att);
    kgat_softmax_b     <<<NK,      B_DOC, 0, stream>>>(ws_att);
    kgat_agg_conv_fused<<<BN / 16, 256,   0, stream>>>(entity_ids, adj_entity, ent_emb,
                                                       ws_att, W_conv, b_conv, out);
}